// FLaGPENodeEncoder_27779848470632
// MI455X (gfx1250) — compile-verified
//
#include <hip/hip_runtime.h>

// ---------------------------------------------------------------------------
// FLaGPE node encoder for MI455X (gfx1250, wave32).
//   out = [ x_out (65536x128) | w_rrwp (256,256,256,8) | delta (256,256,256) ]
// Matrix powers + encoder GEMM run on V_WMMA_F32_16X16X4_F32 (f32-exact).
// Power GEMM: one block = 8 waves = 256x32 output band of one graph.
//   - P column panel (256x32) staged in LDS, transposed with padded stride
//     so B-fragments are single bank-conflict-free ds_load_b64 reads.
//   - 2x2 register tiling per wave (32x32 block, 4 independent WMMA chains).
// Scratch layout: d_ws = [ bufP | bufA ], 2 * 256*256*256 floats = 134 MB.
// Power chain ping-pongs between bufA and the delta region of d_out; delta
// is computed last and overwrites that region.
// ---------------------------------------------------------------------------

typedef float v2f __attribute__((ext_vector_type(2)));
typedef float v8f __attribute__((ext_vector_type(8)));

constexpr int B    = 256;
constexpr int N    = 256;
constexpr int E    = 2048;
constexpr int DIN  = 64;
constexpr int DOUT = 120;   // dim_emb - dim_pe
constexpr int OUTW = 128;   // dim_emb
constexpr int KHOP = 8;
constexpr int NN   = N * N;                 // 65536
constexpr long MAT = (long)B * NN;          // 16,777,216 elems per [B,N,N]
constexpr int KSTR = 260;                   // padded k-stride (words) for LDS panel

// ---------------- zero adjacency scratch -----------------------------------
__global__ void zero_f4_kernel(float4* __restrict__ p, long n4) {
    long i = (long)blockIdx.x * blockDim.x + threadIdx.x;
    if (i < n4) p[i] = make_float4(0.f, 0.f, 0.f, 0.f);
}

// ---------------- scatter-add adjacency ------------------------------------
__global__ void build_adj_kernel(const int* __restrict__ ei, float* __restrict__ adj) {
    int idx = blockIdx.x * blockDim.x + threadIdx.x;   // over B*E
    int b = idx >> 11;          // /E
    int e = idx & (E - 1);
    int s = ei[(b * 2 + 0) * E + e];
    int d = ei[(b * 2 + 1) * E + e];
    atomicAdd(adj + (long)b * NN + s * N + d, 1.0f);
}

// ---------------- row-normalize P = D^-1 A; emit rrwp slots 0 (I) and 1 (P)
__global__ void row_norm_kernel(float* __restrict__ adj, float* __restrict__ wr) {
    int wave = threadIdx.x >> 5;
    int lane = threadIdx.x & 31;
    long row = (long)blockIdx.x * 8 + wave;       // b*N + i
    int  i   = (int)(row & (N - 1));

    float* a = adj + row * N;
    float4 v0 = ((const float4*)a)[lane * 2 + 0];  // j = lane*8 .. lane*8+7
    float4 v1 = ((const float4*)a)[lane * 2 + 1];
    float s = v0.x + v0.y + v0.z + v0.w + v1.x + v1.y + v1.z + v1.w;
    #pragma unroll
    for (int m = 16; m >= 1; m >>= 1) s += __shfl_xor(s, m, 32);
    float inv = 1.0f / fmaxf(s, 1.0f);            // deg = clip(sum, 1, inf)

    float val[8];
    ((float4*)val)[0] = v0;
    ((float4*)val)[1] = v1;
    float* wrow = wr + row * N * KHOP;            // ((b*N+i)*N + j)*8
    #pragma unroll
    for (int q = 0; q < 8; ++q) {
        int j = lane * 8 + q;
        float p = val[q] * inv;
        a[j] = p;                                  // P in place
        wrow[j * KHOP + 0] = (j == i) ? 1.f : 0.f; // slot 0: identity
        wrow[j * KHOP + 1] = p;                    // slot 1: P
    }
}

// ---------------- A_{t+1} = A_t * P  (LDS-staged P panel, 2x2 WMMA tiling) --
__global__ __launch_bounds__(256)
void rrwp_power_gemm_kernel(const float* __restrict__ A,
                            const float* __restrict__ Pm,
                            float* __restrict__ Anext,
                            float* __restrict__ wr, int kslot) {
    __shared__ float panel[32 * KSTR];  // P panel transposed: panel[c][k]

    int wave = threadIdx.x >> 5;
    int lane = threadIdx.x & 31;
    int b  = blockIdx.x >> 3;           // graph
    int jt = blockIdx.x & 7;            // 32-wide column band

    const float* Pb = Pm + (long)b * NN;

    // Cooperative fill: wave w loads rows k = w, w+8, ... (coalesced 128B rows),
    // stores transposed so B-fragments read contiguous (k, k+1) pairs.
    for (int k = wave; k < N; k += 8) {
        panel[lane * KSTR + k] = Pb[k * N + jt * 32 + lane];
    }
    __syncthreads();

    int it = wave;                      // 32-row band within the graph
    int lm = lane & 15;
    int hl = lane >> 4;
    const float* a0row = A + (long)b * NN + (it * 32 + lm) * N;
    const float* a1row = a0row + 16 * N;
    const float* bp0 = &panel[lm * KSTR];          // cols jt*32 + lm
    const float* bp1 = &panel[(lm + 16) * KSTR];   // cols jt*32 + 16 + lm

    v8f acc00 = {}, acc01 = {}, acc10 = {}, acc11 = {};
    #pragma unroll 4
    for (int kk = 0; kk < 64; ++kk) {
        int k0 = kk * 4 + 2 * hl;
        float2 a0 = *(const float2*)(a0row + k0);  // A[i0+lm][k0..k0+1]
        float2 a1 = *(const float2*)(a1row + k0);  // A[i0+16+lm][k0..k0+1]
        float2 b0 = *(const float2*)(bp0 + k0);    // P[k0..k0+1][c0]
        float2 b1 = *(const float2*)(bp1 + k0);    // P[k0..k0+1][c1]
        v2f av0; av0.x = a0.x; av0.y = a0.y;
        v2f av1; av1.x = a1.x; av1.y = a1.y;
        v2f bv0; bv0.x = b0.x; bv0.y = b0.y;
        v2f bv1; bv1.x = b1.x; bv1.y = b1.y;
        acc00 = __builtin_amdgcn_wmma_f32_16x16x4_f32(false, av0, false, bv0, (short)0, acc00, false, false);
        acc01 = __builtin_amdgcn_wmma_f32_16x16x4_f32(false, av0, false, bv1, (short)0, acc01, false, false);
        acc10 = __builtin_amdgcn_wmma_f32_16x16x4_f32(false, av1, false, bv0, (short)0, acc10, false, false);
        acc11 = __builtin_amdgcn_wmma_f32_16x16x4_f32(false, av1, false, bv1, (short)0, acc11, false, false);
    }

    float* ob  = Anext + (long)b * NN;
    float* wrb = wr + (long)b * NN * KHOP + kslot;
    int c0 = jt * 32 + lm;
    int c1 = c0 + 16;
    #pragma unroll
    for (int r = 0; r < 8; ++r) {
        int i0 = it * 32 + r + 8 * hl;  // C layout: VGPR r -> row r / r+8
        int i1 = i0 + 16;
        ob[i0 * N + c0] = acc00[r];
        ob[i0 * N + c1] = acc01[r];
        ob[i1 * N + c0] = acc10[r];
        ob[i1 * N + c1] = acc11[r];
        wrb[((long)i0 * N + c0) * KHOP] = acc00[r];
        wrb[((long)i0 * N + c1) * KHOP] = acc01[r];
        wrb[((long)i1 * N + c0) * KHOP] = acc10[r];
        wrb[((long)i1 * N + c1) * KHOP] = acc11[r];
    }
}

// ---------------- delta_ij[b,i,j] = (frag[b,i] == frag[b,j]) ----------------
__global__ void frag_delta_kernel(const int* __restrict__ fid, float* __restrict__ delta) {
    int bi = blockIdx.x;           // b*N + i
    int b  = bi >> 8;
    int fi = fid[bi];
    int j  = threadIdx.x;
    int fj = fid[b * N + j];
    delta[(long)bi * N + j] = (fi == fj) ? 1.f : 0.f;
}

// ---------------- x_out = [x @ W + b | zeros(8)]  (WMMA f32) ----------------
__global__ void node_encoder_kernel(const float* __restrict__ X,
                                    const float* __restrict__ W,
                                    const float* __restrict__ bias,
                                    float* __restrict__ out) {
    int wave = threadIdx.x >> 5;
    int lane = threadIdx.x & 31;
    int tile = blockIdx.x * 8 + wave;   // (BN/16) row-tiles x 8 col-tiles
    int mt = tile >> 3;
    int jt = tile & 7;
    int hl = lane >> 4;
    int lm = lane & 15;

    const float* xrow = X + (long)(mt * 16 + lm) * DIN;
    int col = jt * 16 + lm;

    v8f acc = {};
    #pragma unroll
    for (int kk = 0; kk < 16; ++kk) {
        int k0 = kk * 4 + 2 * hl;
        float2 ald = *(const float2*)(xrow + k0);
        v2f av; av.x = ald.x; av.y = ald.y;
        v2f bv;
        bv.x = (col < DOUT) ? W[(k0 + 0) * DOUT + col] : 0.f;
        bv.y = (col < DOUT) ? W[(k0 + 1) * DOUT + col] : 0.f;
        acc = __builtin_amdgcn_wmma_f32_16x16x4_f32(
            false, av, false, bv, (short)0, acc, false, false);
    }
    float bb = (col < DOUT) ? bias[col] : 0.f;
    #pragma unroll
    for (int r = 0; r < 8; ++r) {
        int i = mt * 16 + r + 8 * hl;
        out[(long)i * OUTW + col] = (col < DOUT) ? (acc[r] + bb) : 0.f;
    }
}

// ---------------------------------------------------------------------------
extern "C" void kernel_launch(void* const* d_in, const int* in_sizes, int n_in,
                              void* d_out, int out_size, void* d_ws, size_t ws_size,
                              hipStream_t stream) {
    (void)in_sizes; (void)n_in; (void)out_size; (void)ws_size;

    const float* x    = (const float*)d_in[0];   // [B*N, DIN]
    const float* W    = (const float*)d_in[1];   // [DIN, DOUT]
    const float* bx   = (const float*)d_in[2];   // [DOUT]
    const int*   ei   = (const int*)  d_in[3];   // [B, 2, E]
    const int*   fid  = (const int*)  d_in[4];   // [B, N]

    float* out   = (float*)d_out;
    float* xout  = out;
    float* wr    = out + (long)B * N * OUTW;     // + 8,388,608
    float* delta = wr  + MAT * KHOP;             // + 134,217,728

    float* bufP = (float*)d_ws;                  // P (fixed right operand)
    float* bufA = bufP + MAT;                    // power ping buffer

    // 1) adjacency -> P
    zero_f4_kernel<<<(int)(MAT / 4 / 256), 256, 0, stream>>>((float4*)bufP, MAT / 4);
    build_adj_kernel<<<(B * E) / 256, 256, 0, stream>>>(ei, bufP);
    row_norm_kernel<<<(B * N) / 8, 256, 0, stream>>>(bufP, wr);

    // 2) P^2 .. P^7, ping-ponging bufA <-> delta region
    const float* src = bufP;                     // A_1 = P
    float* pp[2] = { bufA, delta };
    for (int t = 2; t <= 7; ++t) {
        float* dst = pp[t & 1];                  // t=2->bufA, t=3->delta, ...
        rrwp_power_gemm_kernel<<<B * 8, 256, 0, stream>>>(src, bufP, dst, wr, t);
        src = dst;
    }

    // 3) fragment mask (overwrites delta region last)
    frag_delta_kernel<<<B * N, 256, 0, stream>>>(fid, delta);

    // 4) node features
    node_encoder_kernel<<<((B * N / 16) * 8) / 8, 256, 0, stream>>>(x, W, bx, xout);
}